// SinkhornDistance_84189948936349
// MI455X (gfx1250) — compile-verified
//
#include <hip/hip_runtime.h>

#define P        2048
#define NBATCH   8
#define EPSV     0.1f
#define INV_EPS  10.0f
#define MAX_ITER 50
#define EPS_LOG  1e-8f
#define TPB      1024
#define NWAVES   (TPB / 32)
#define NTILE    (P / 16)        // 128 tiles per dimension

typedef __attribute__((ext_vector_type(2))) float v2f;
typedef __attribute__((ext_vector_type(8))) float v8f;

__global__ __launch_bounds__(TPB, 1)
void SinkhornDistance_84189948936349_kernel(const float* __restrict__ x,
                                            const float* __restrict__ y,
                                            float* __restrict__ out) {
  __shared__ float sx0[P], sx1[P], sy0[P], sy1[P], su[P], sv[P];
  __shared__ float redA[NWAVES], redB[NWAVES];
  __shared__ float sSx, sSy;

  const int b    = blockIdx.x;
  const int tid  = threadIdx.x;
  const int lane = tid & 31;
  const int wave = tid >> 5;
  const int half = lane >> 4;     // 0: rows v+0..7 / K=0,1 ; 1: rows v+8..15 / K=2,3
  const int l15  = lane & 15;

  const float* xb = x + (size_t)b * P * 2;
  const float* yb = y + (size_t)b * P * 2;

  // ---- load, mask (mask from x applies to both x and y per reference), marginal sums ----
  float px = 0.f, py = 0.f;
  for (int i = tid; i < P; i += TPB) {
    float x0 = xb[2 * i], x1 = xb[2 * i + 1];
    float y0 = yb[2 * i], y1 = yb[2 * i + 1];
    float m0 = (x0 != -1.f) ? 1.f : 0.f;
    float m1 = (x1 != -1.f) ? 1.f : 0.f;
    x0 *= m0; x1 *= m1; y0 *= m0; y1 *= m1;
    sx0[i] = x0; sx1[i] = x1; sy0[i] = y0; sy1[i] = y1;
    su[i] = 0.f; sv[i] = 0.f;
    px += x1 * x1; py += y1 * y1;
  }
  #pragma unroll
  for (int k = 16; k >= 1; k >>= 1) {
    px += __shfl_xor(px, k, 32);
    py += __shfl_xor(py, k, 32);
  }
  if (lane == 0) { redA[wave] = px; redB[wave] = py; }
  __syncthreads();
  if (tid == 0) {
    float sx = 0.f, sy = 0.f;
    for (int w = 0; w < NWAVES; ++w) { sx += redA[w]; sy += redB[w]; }
    sSx = sx; sSy = sy;
  }
  __syncthreads();
  const float Sx = sSx, Sy = sSy;

  // One V_WMMA_F32_16X16X4_F32 builds a full 16x16 cost tile:
  //   A row r = (-2x0, -2x1, |x_r|^2, 1)   (K=0,1 in lanes 0-15; K=2,3 in lanes 16-31)
  //   B col c = ( y0 ,  y1 ,   1 , |y_c|^2)
  //   D = A*B + 0  =  |x_r|^2 + |y_c|^2 - 2 x_r . y_c  =  C[r][c]
  auto cost_tile = [&](int i0, int j0) -> v8f {
    float xr0 = sx0[i0 + l15], xr1 = sx1[i0 + l15];
    float yc0 = sy0[j0 + l15], yc1 = sy1[j0 + l15];
    v2f a, bb;
    a.x  = half ? (xr0 * xr0 + xr1 * xr1) : (-2.f * xr0);
    a.y  = half ? 1.f                     : (-2.f * xr1);
    bb.x = half ? 1.f                     : yc0;
    bb.y = half ? (yc0 * yc0 + yc1 * yc1) : yc1;
    v8f c = {};
    return __builtin_amdgcn_wmma_f32_16x16x4_f32(false, a, false, bb,
                                                 (short)0, c, false, false);
  };

  // ---- 50 Sinkhorn iterations, everything LDS-resident ----
  for (int it = 0; it < MAX_ITER; ++it) {
    // u-update: per row r, L_r = logsumexp_j((v_j - C_rj)/eps); u_r = eps*(logmu_r - L_r)
    for (int rt = wave; rt < NTILE; rt += NWAVES) {
      int i0 = rt << 4;
      float m[8];
      #pragma unroll
      for (int v = 0; v < 8; ++v) m[v] = -3.0e38f;
      for (int jt = 0; jt < NTILE; ++jt) {          // max pass
        v8f d = cost_tile(i0, jt << 4);
        float vc = sv[(jt << 4) + l15];
        #pragma unroll
        for (int v = 0; v < 8; ++v)
          m[v] = fmaxf(m[v], (vc - d[v]) * INV_EPS);
      }
      #pragma unroll
      for (int k = 1; k < 16; k <<= 1)
        #pragma unroll
        for (int v = 0; v < 8; ++v)
          m[v] = fmaxf(m[v], __shfl_xor(m[v], k, 32));
      float s[8];
      #pragma unroll
      for (int v = 0; v < 8; ++v) s[v] = 0.f;
      for (int jt = 0; jt < NTILE; ++jt) {          // sum pass: 1 exp per element
        v8f d = cost_tile(i0, jt << 4);
        float vc = sv[(jt << 4) + l15];
        #pragma unroll
        for (int v = 0; v < 8; ++v)
          s[v] += __expf((vc - d[v]) * INV_EPS - m[v]);
      }
      #pragma unroll
      for (int k = 1; k < 16; k <<= 1)
        #pragma unroll
        for (int v = 0; v < 8; ++v)
          s[v] += __shfl_xor(s[v], k, 32);
      if (l15 == 0) {                               // lanes 0 & 16 write 8 rows each
        #pragma unroll
        for (int v = 0; v < 8; ++v) {
          int r = i0 + v + 8 * half;
          float x1r = sx1[r];
          float logmu = __logf(x1r * x1r / Sx + EPS_LOG);
          su[r] = EPSV * (logmu - (m[v] + __logf(s[v])));
        }
      }
    }
    __syncthreads();
    // v-update: per col c, L_c = logsumexp_i((u_i - C_ic)/eps); v_c = eps*(lognu_c - L_c)
    for (int ct = wave; ct < NTILE; ct += NWAVES) {
      int j0 = ct << 4;
      float m = -3.0e38f;
      for (int itl = 0; itl < NTILE; ++itl) {       // max pass
        int i0 = itl << 4;
        v8f d = cost_tile(i0, j0);
        #pragma unroll
        for (int v = 0; v < 8; ++v)
          m = fmaxf(m, (su[i0 + v + 8 * half] - d[v]) * INV_EPS);
      }
      m = fmaxf(m, __shfl_xor(m, 16, 32));          // merge row halves
      float s = 0.f;
      for (int itl = 0; itl < NTILE; ++itl) {       // sum pass
        int i0 = itl << 4;
        v8f d = cost_tile(i0, j0);
        #pragma unroll
        for (int v = 0; v < 8; ++v)
          s += __expf((su[i0 + v + 8 * half] - d[v]) * INV_EPS - m);
      }
      s += __shfl_xor(s, 16, 32);
      if (half == 0) {
        int c = j0 + l15;
        float y1c = sy1[c];
        float lognu = __logf(y1c * y1c / Sy + EPS_LOG);
        sv[c] = EPSV * (lognu - (m + __logf(s)));
      }
    }
    __syncthreads();
  }

  // ---- emit pi, C, cost ----
  const size_t PP = (size_t)P * P;
  float* outPi = out + NBATCH + (size_t)b * PP;
  float* outC  = out + NBATCH + (size_t)NBATCH * PP + (size_t)b * PP;
  float costacc = 0.f;
  for (int t = wave; t < NTILE * NTILE; t += NWAVES) {
    int i0 = (t >> 7) << 4;
    int j0 = (t & 127) << 4;
    v8f d = cost_tile(i0, j0);
    float vc = sv[j0 + l15];
    #pragma unroll
    for (int v = 0; v < 8; ++v) {
      int r = i0 + v + 8 * half;
      float C  = d[v];
      float pi = __expf((su[r] + vc - C) * INV_EPS);
      size_t off = (size_t)r * P + (size_t)(j0 + l15);
      outC[off]  = C;
      outPi[off] = pi;
      costacc += pi * C;
    }
  }
  #pragma unroll
  for (int k = 16; k >= 1; k >>= 1) costacc += __shfl_xor(costacc, k, 32);
  if (lane == 0) redA[wave] = costacc;
  __syncthreads();
  if (tid == 0) {
    float c = 0.f;
    for (int w = 0; w < NWAVES; ++w) c += redA[w];  // fixed-order: deterministic
    out[b] = c;
  }
}

extern "C" void kernel_launch(void* const* d_in, const int* in_sizes, int n_in,
                              void* d_out, int out_size, void* d_ws, size_t ws_size,
                              hipStream_t stream) {
  (void)in_sizes; (void)n_in; (void)d_ws; (void)ws_size; (void)out_size;
  const float* x = (const float*)d_in[0];
  const float* y = (const float*)d_in[1];
  float* out = (float*)d_out;
  SinkhornDistance_84189948936349_kernel<<<dim3(NBATCH), dim3(TPB), 0, stream>>>(x, y, out);
}